// EntropyPool2d_62191126446160
// MI455X (gfx1250) — compile-verified
//
#include <hip/hip_runtime.h>
#include <stdint.h>

// EntropyPool2d, MAX_ENTROPY variant, K=3, stride 1, valid padding.
// x: (8,32,256,256) fp32 holding integer values 0..255.
// out: (8,32,254,254) fp32.
//
// Pass 1: global 256-bin histogram of the quantized values.
// Pass 2: per 3x3 window, emit the value whose global count is minimal
//         (first minimum in row-major (di,dj) order).

#define KS 3
#define H  256
#define W  256
#define HO 254
#define WO 254
#define TH 30                                  // output rows per block
#define STRIPS ((HO + TH - 1) / TH)            // 9
#define NROW (TH + 2)                          // 32 input rows staged
#define NV4  (NROW * (W / 4))                  // 2048 float4 loads per tile

#define NSUB 8                                 // one sub-histogram per wave32
#define SUBSTRIDE 260                          // pad: copies start 4 banks apart

__global__ __launch_bounds__(256)
void zero_hist_kernel(int* __restrict__ ghist) {
    ghist[threadIdx.x] = 0;
}

__global__ __launch_bounds__(256)
void hist_kernel(const float* __restrict__ x, int* __restrict__ ghist, int n4) {
    __shared__ int lh[NSUB * SUBSTRIDE];
    const int tid = threadIdx.x;
    for (int i = tid; i < NSUB * SUBSTRIDE; i += 256) lh[i] = 0;
    __syncthreads();

    // Each wave32 owns a private copy: no inter-wave LDS-atomic contention.
    int* __restrict__ myh = &lh[(tid >> 5) * SUBSTRIDE];

    const float4* __restrict__ x4 = (const float4*)x;
    for (int i = blockIdx.x * blockDim.x + tid; i < n4; i += gridDim.x * blockDim.x) {
        float4 v = x4[i];
        atomicAdd(&myh[((int)v.x) & 255], 1);
        atomicAdd(&myh[((int)v.y) & 255], 1);
        atomicAdd(&myh[((int)v.z) & 255], 1);
        atomicAdd(&myh[((int)v.w) & 255], 1);
    }
    __syncthreads();

    int c = 0;
    #pragma unroll
    for (int s = 0; s < NSUB; ++s) c += lh[s * SUBSTRIDE + tid];
    if (c) atomicAdd(&ghist[tid], c);
}

__global__ __launch_bounds__(256)
void pool_kernel(const float* __restrict__ x, const int* __restrict__ ghist,
                 float* __restrict__ out) {
    __shared__ __align__(16) float tile[NROW * W];   // 32 KB staged input rows
    __shared__ int lh[256];                          // 1 KB global-count table

    const int tid   = threadIdx.x;
    const int p     = blockIdx.x / STRIPS;           // (b,c) plane index
    const int strip = blockIdx.x % STRIPS;
    const int r0    = strip * TH;

    // Count table: plain load (tiny, L2-hot).
    lh[tid] = ghist[tid];

    // Stage 32 input rows (row-clamped for the ragged last strip) into LDS
    // with gfx1250 async copies: 8x b128 per thread, tracked by ASYNCcnt.
    const float* __restrict__ plane = x + (size_t)p * (H * W);
    for (int k = tid; k < NV4; k += 256) {
        int row  = k >> 6;          // 64 float4 per row
        int c4   = k & 63;
        int grow = r0 + row;
        if (grow > H - 1) grow = H - 1;
        unsigned lds_addr = (unsigned)(uintptr_t)(&tile[row * W + c4 * 4]);
        unsigned long long gaddr =
            (unsigned long long)(uintptr_t)(plane + grow * W + c4 * 4);
        asm volatile("global_load_async_to_lds_b128 %0, %1, off"
                     :: "v"(lds_addr), "v"(gaddr) : "memory");
    }
    asm volatile("s_wait_asynccnt 0" ::: "memory");
    __syncthreads();

    // Each thread owns one output column for TH rows.
    if (tid < WO) {
        float* __restrict__ oplane = out + (size_t)p * (HO * WO);
        for (int y = 0; y < TH; ++y) {
            int oy = r0 + y;
            if (oy >= HO) break;
            int   best = 0x7fffffff;
            float bval = 0.0f;
            // Row-major (di,dj) scan with strict '<' => first-minimum
            // tie-break, matching jnp.argmin / torch.min semantics.
            #pragma unroll
            for (int di = 0; di < KS; ++di) {
                #pragma unroll
                for (int dj = 0; dj < KS; ++dj) {
                    float v = tile[(y + di) * W + tid + dj];
                    int   c = lh[((int)v) & 255];
                    if (c < best) { best = c; bval = v; }
                }
            }
            // Output is written once and never re-read: keep it out of L2.
            __builtin_nontemporal_store(bval, &oplane[oy * WO + tid]);
        }
    }
}

extern "C" void kernel_launch(void* const* d_in, const int* in_sizes, int n_in,
                              void* d_out, int out_size, void* d_ws, size_t ws_size,
                              hipStream_t stream) {
    const float* x   = (const float*)d_in[0];
    float*       out = (float*)d_out;
    int*         gh  = (int*)d_ws;              // 256 ints of scratch

    const int n      = in_sizes[0];             // 8*32*256*256 = 16,777,216
    const int planes = n / (H * W);             // 256

    zero_hist_kernel<<<1, 256, 0, stream>>>(gh);
    hist_kernel<<<1024, 256, 0, stream>>>(x, gh, n / 4);
    pool_kernel<<<planes * STRIPS, 256, 0, stream>>>(x, gh, out);
}